// UniformLoss_61194694033710
// MI455X (gfx1250) — compile-verified
//
#include <hip/hip_runtime.h>
#include <math.h>

#define NBATCH 8
#define NPTS   8192
#define NPOINT 409                 // int(8192*0.05)
#define NGROUP (NBATCH*NPOINT)     // 3272

typedef __attribute__((ext_vector_type(2))) float v2f;
typedef __attribute__((ext_vector_type(8))) float v8f;

// ---------------------------------------------------------------------------
// Kernel 1: farthest point sampling. One 1024-thread workgroup per batch.
// Each thread keeps 8 points (coords + running min dist) entirely in VGPRs.
// Argmax with first-occurrence tie-break (explicit index compare) to match
// jnp.argmax semantics.
// ---------------------------------------------------------------------------
__global__ __launch_bounds__(1024)
void fps_kernel(const float* __restrict__ pcd, int* __restrict__ fps_idx) {
  const int b    = blockIdx.x;
  const int tid  = threadIdx.x;
  const int lane = tid & 31;
  const int wv   = tid >> 5;
  const float* base = pcd + (size_t)b * NPTS * 3;

  float px[8], py[8], pz[8], dd[8];
#pragma unroll
  for (int k = 0; k < 8; ++k) {
    const int j = k * 1024 + tid;
    px[k] = base[j * 3 + 0];
    py[k] = base[j * 3 + 1];
    pz[k] = base[j * 3 + 2];
    dd[k] = 1e10f;
  }

  __shared__ float sv[32], sx[32], sy[32], sz[32];
  __shared__ int   si[32];
  __shared__ float bx, by, bz;
  __shared__ int   bi;

  int   cur = 0;
  float cx = base[0], cy = base[1], cz = base[2];

  for (int s = 0; s < NPOINT; ++s) {
    if (tid == 0) fps_idx[b * NPOINT + s] = cur;

    float bestv = -1.0f, bxx = 0.f, byy = 0.f, bzz = 0.f;
    int   besti = 0;
#pragma unroll
    for (int k = 0; k < 8; ++k) {
      const float dx = px[k] - cx, dy = py[k] - cy, dz = pz[k] - cz;
      const float d  = dx * dx + dy * dy + dz * dz;
      const float nd = fminf(dd[k], d);
      dd[k] = nd;
      const int j = k * 1024 + tid;
      if (nd > bestv || (nd == bestv && j < besti)) {
        bestv = nd; besti = j; bxx = px[k]; byy = py[k]; bzz = pz[k];
      }
    }
    // wave32 reduction
    for (int off = 16; off > 0; off >>= 1) {
      const float ov = __shfl_down(bestv, off, 32);
      const int   oi = __shfl_down(besti, off, 32);
      const float ox = __shfl_down(bxx,   off, 32);
      const float oy = __shfl_down(byy,   off, 32);
      const float oz = __shfl_down(bzz,   off, 32);
      if (ov > bestv || (ov == bestv && oi < besti)) {
        bestv = ov; besti = oi; bxx = ox; byy = oy; bzz = oz;
      }
    }
    if (lane == 0) { sv[wv] = bestv; si[wv] = besti; sx[wv] = bxx; sy[wv] = byy; sz[wv] = bzz; }
    __syncthreads();
    if (tid < 32) {
      bestv = sv[tid]; besti = si[tid]; bxx = sx[tid]; byy = sy[tid]; bzz = sz[tid];
      for (int off = 16; off > 0; off >>= 1) {
        const float ov = __shfl_down(bestv, off, 32);
        const int   oi = __shfl_down(besti, off, 32);
        const float ox = __shfl_down(bxx,   off, 32);
        const float oy = __shfl_down(byy,   off, 32);
        const float oz = __shfl_down(bzz,   off, 32);
        if (ov > bestv || (ov == bestv && oi < besti)) {
          bestv = ov; besti = oi; bxx = ox; byy = oy; bzz = oz;
        }
      }
      if (tid == 0) { bi = besti; bx = bxx; by = byy; bz = bzz; }
    }
    __syncthreads();
    cur = bi; cx = bx; cy = by; cz = bz;
  }
}

// ---------------------------------------------------------------------------
// Fully-unrolled row-tile NN search via V_WMMA_F32_16X16X4_F32 Gram tiles.
// T = #16-wide column tiles (compile-time), NSAMP = real sample count.
// Called with wave-uniform control flow => EXEC all-ones (WMMA requirement).
// ---------------------------------------------------------------------------
template<int T, int NSAMP>
__device__ __forceinline__ void nn_rowtile(const float4* __restrict__ pts,
                                           int ti, int lane,
                                           float* __restrict__ taskOut) {
  const int m  = lane & 15;
  const int hi = lane >> 4;

  // A operand: 16x4 f32 layout — lanes 0-15 hold (K0,K1), lanes 16-31 (K2,K3)
  const float4 pa = pts[ti * 16 + m];
  v2f a; a.x = hi ? pa.z : pa.x; a.y = hi ? 0.f : pa.y;

  float nrow[8];
#pragma unroll
  for (int r = 0; r < 8; ++r)
    nrow[r] = pts[ti * 16 + hi * 8 + r].w;

  float acc[8];
#pragma unroll
  for (int r = 0; r < 8; ++r) acc[r] = 3.4e38f;

#pragma unroll
  for (int tj = 0; tj < T; ++tj) {
    const float4 pb = pts[tj * 16 + m];
    v2f bo; bo.x = hi ? pb.z : pb.x; bo.y = hi ? 0.f : pb.y;
    v8f c = {0.f, 0.f, 0.f, 0.f, 0.f, 0.f, 0.f, 0.f};
    // G tile = A * A^T  (B operand layout coincides with A for the transpose)
    c = __builtin_amdgcn_wmma_f32_16x16x4_f32(
            false, a, false, bo, (short)0, c, false, false);
    const float ncol = pb.w;
    const int   col  = tj * 16 + m;
#pragma unroll
    for (int r = 0; r < 8; ++r) {
      const int row = ti * 16 + hi * 8 + r;   // C/D layout: VGPR r, halves +8
      float d2 = nrow[r] + ncol - 2.0f * c[r];
      if (row == col) d2 = 3.4e38f;           // exclude self (d2[i][i]=0)
      acc[r] = fminf(acc[r], d2);
    }
  }
  // min across the 16 lanes of each half (they share the same 8 rows)
#pragma unroll
  for (int off = 1; off < 16; off <<= 1) {
#pragma unroll
    for (int r = 0; r < 8; ++r)
      acc[r] = fminf(acc[r], __shfl_xor(acc[r], off, 32));
  }
  if (m == 0) {
    float ssum = 0.f;
#pragma unroll
    for (int r = 0; r < 8; ++r) {
      const int row = ti * 16 + hi * 8 + r;
      if (row < NSAMP) ssum += sqrtf(fmaxf(acc[r], 0.f));
    }
    taskOut[hi] = ssum;
  }
}

// ---------------------------------------------------------------------------
// Kernel 2: fused ball-query + per-group uniform loss. One block per
// (batch, center). Waves 0..4 scan one radius each (index-order selection via
// ballot prefix), then all 8 waves run WMMA row-tile tasks.
// ---------------------------------------------------------------------------
__global__ __launch_bounds__(256)
void group_kernel(const float* __restrict__ pcd, const int* __restrict__ fps_idx,
                  float* __restrict__ blockSums) {
  constexpr int    NS[5]    = {32, 49, 65, 81, 98};
  constexpr int    TILES[5] = {2, 4, 5, 6, 7};
  constexpr int    START[5] = {0, 2, 6, 11, 17};     // 24 row-tile tasks total
  constexpr int    TASK_PC[24] = {0,0, 1,1,1,1, 2,2,2,2,2,
                                  3,3,3,3,3,3, 4,4,4,4,4,4,4};
  constexpr double PCT[5]   = {0.004, 0.006, 0.008, 0.01, 0.012};

  __shared__ float4 spts[5][112];   // per-percentage neighbor coords + |x|^2
  __shared__ float  taskSum[24][2];

  const int g    = blockIdx.x;
  const int b    = g / NPOINT;
  const int tid  = threadIdx.x;
  const int lane = tid & 31;
  const int wv   = tid >> 5;
  const float* base = pcd + (size_t)b * NPTS * 3;

  const int   cidx = fps_idx[g];
  const float cx = base[cidx * 3 + 0];
  const float cy = base[cidx * 3 + 1];
  const float cz = base[cidx * 3 + 2];

  // -------- phase 1: ball query (waves 0..4, one radius each) --------
  if (wv < 5) {
    const int ns = NS[wv];
    const double rd = sqrt(PCT[wv]);
    const float thr = (float)(rd * rd);          // matches python r*r
    int cnt = 0;
    float4 f0 = make_float4(0.f, 0.f, 0.f, 0.f); // first in-ball point (cnt>=1 always)
    for (int it = 0; it < NPTS / 32 && cnt < ns; ++it) {
      const int   j  = it * 32 + lane;
      const float x  = base[j * 3 + 0];
      const float y  = base[j * 3 + 1];
      const float z  = base[j * 3 + 2];
      const float dx = x - cx, dy = y - cy, dz = z - cz;
      const float d2 = dx * dx + dy * dy + dz * dz;
      const bool  sel = d2 < thr;
      const unsigned mask = __builtin_amdgcn_ballot_w32(sel);
      const int pre  = __popc(mask & ((1u << lane) - 1u));
      const int slot = cnt + pre;
      if (sel && slot < ns)
        spts[wv][slot] = make_float4(x, y, z, x * x + y * y + z * z);
      if (cnt == 0 && mask != 0u) {
        const int src = __ffs((int)mask) - 1;
        f0.x = __shfl(x, src, 32);
        f0.y = __shfl(y, src, 32);
        f0.z = __shfl(z, src, 32);
        f0.w = f0.x * f0.x + f0.y * f0.y + f0.z * f0.z;
      }
      cnt += __popc(mask);
      if (cnt > ns) cnt = ns;
    }
    // pad unfilled sample slots with first neighbor (reference semantics)
    for (int sl = cnt + lane; sl < ns; sl += 32) spts[wv][sl] = f0;
    // pad to tile multiple with far-away sentinel points
    const float4 sent = make_float4(1e6f, 1e6f, 1e6f, 3e12f);
    for (int sl = ns + lane; sl < TILES[wv] * 16; sl += 32) spts[wv][sl] = sent;
  }
  __syncthreads();

  // -------- phase 2: WMMA row-tile tasks (wave-uniform dispatch) --------
  for (int task = wv; task < 24; task += 8) {
    const int pc = TASK_PC[task];
    const int ti = task - START[pc];
    switch (pc) {
      case 0: nn_rowtile<2, 32>(spts[0], ti, lane, taskSum[task]); break;
      case 1: nn_rowtile<4, 49>(spts[1], ti, lane, taskSum[task]); break;
      case 2: nn_rowtile<5, 65>(spts[2], ti, lane, taskSum[task]); break;
      case 3: nn_rowtile<6, 81>(spts[3], ti, lane, taskSum[task]); break;
      default: nn_rowtile<7, 98>(spts[4], ti, lane, taskSum[task]); break;
    }
  }
  __syncthreads();

  // -------- phase 3: deterministic per-group loss values --------
  if (tid < 5) {
    const int pc = tid;
    float ssum = 0.f;
    for (int ti = 0; ti < TILES[pc]; ++ti)
      ssum += taskSum[START[pc] + ti][0] + taskSum[START[pc] + ti][1];
    const float mean   = ssum / (float)NS[pc] + 1e-8f;
    const float expect = (float)sqrt(3.14159265358979323846 / (double)NPTS);
    const float diff   = mean - expect;
    blockSums[(size_t)g * 5 + pc] = diff * diff / (expect + 1e-8f);
  }
}

// ---------------------------------------------------------------------------
// Kernel 3: deterministic single-block reduction -> scalar loss.
// ---------------------------------------------------------------------------
__global__ __launch_bounds__(256)
void final_kernel(const float* __restrict__ blockSums, float* __restrict__ out) {
  __shared__ float red[256];
  const int tid = threadIdx.x;
  constexpr double PCT[5] = {0.004, 0.006, 0.008, 0.01, 0.012};
  float loss = 0.f;
  for (int pc = 0; pc < 5; ++pc) {
    float s = 0.f;
    for (int gi = tid; gi < NGROUP; gi += 256)
      s += blockSums[(size_t)gi * 5 + pc];
    red[tid] = s;
    __syncthreads();
    for (int off = 128; off > 0; off >>= 1) {
      if (tid < off) red[tid] += red[tid + off];
      __syncthreads();
    }
    if (tid == 0) {
      const float w = (float)((PCT[pc] * 100.0) * (PCT[pc] * 100.0));
      loss += (red[0] / (float)NGROUP) * w;
    }
    __syncthreads();
  }
  if (tid == 0) out[0] = loss / 5.0f * (float)NBATCH;
}

// ---------------------------------------------------------------------------
extern "C" void kernel_launch(void* const* d_in, const int* in_sizes, int n_in,
                              void* d_out, int out_size, void* d_ws, size_t ws_size,
                              hipStream_t stream) {
  (void)in_sizes; (void)n_in; (void)out_size; (void)ws_size;
  const float* pcd = (const float*)d_in[0];
  float* out = (float*)d_out;
  char*  ws  = (char*)d_ws;

  int*   fps_idx   = (int*)ws;                                        // 3272 ints
  float* blockSums = (float*)(ws + ((NGROUP * sizeof(int) + 255) & ~(size_t)255));
                                                                      // 16360 floats

  fps_kernel  <<<NBATCH, 1024, 0, stream>>>(pcd, fps_idx);
  group_kernel<<<NGROUP,  256, 0, stream>>>(pcd, fps_idx, blockSums);
  final_kernel<<<1,       256, 0, stream>>>(blockSums, out);
}